// FeaturePropagation_54219667144942
// MI455X (gfx1250) — compile-verified
//
#include <hip/hip_runtime.h>

typedef float v2f __attribute__((ext_vector_type(2)));
typedef float v8f __attribute__((ext_vector_type(8)));

#define WMMA_F32(a, b, c) \
    __builtin_amdgcn_wmma_f32_16x16x4_f32(false, (a), false, (b), (short)0, (c), false, false)

// ---------------------------------------------------------------------------
// constants from the reference
// ---------------------------------------------------------------------------
#define BB   4
#define NN   16384
#define SS   4096
#define D1C  128
#define D2C  256
#define C0K  384    // D1 + D2
#define COUT 256
#define EPSB 1e-5f

// ---------------------------------------------------------------------------
// Kernel 1: exhaustive 3-NN (streamed, xyz2 cached in LDS)
// ---------------------------------------------------------------------------
__global__ __launch_bounds__(256)
void knn3_kernel(const float* __restrict__ xyz1, const float* __restrict__ xyz2,
                 int* __restrict__ idx_out, float* __restrict__ w_out)
{
    __shared__ float px[SS], py[SS], pz[SS];   // 48 KB
    const int b = blockIdx.y;

    for (int s = threadIdx.x; s < SS; s += 256) {
        const float* p = xyz2 + ((size_t)b * SS + s) * 3;
        px[s] = p[0]; py[s] = p[1]; pz[s] = p[2];
    }
    __syncthreads();

    const int n = blockIdx.x * 256 + threadIdx.x;
    const float* q = xyz1 + ((size_t)b * NN + n) * 3;
    const float qx = q[0], qy = q[1], qz = q[2];
    const float qn = qx*qx + qy*qy + qz*qz;

    float d0 = 3.4e38f, d1 = 3.4e38f, d2 = 3.4e38f;
    int   i0 = 0, i1 = 0, i2 = 0;

    #pragma unroll 4
    for (int s = 0; s < SS; ++s) {
        const float x = px[s], y = py[s], z = pz[s];
        const float pn = x*x + y*y + z*z;
        const float dot = qx*x + qy*y + qz*z;
        const float d = qn + pn - 2.0f * dot;      // same formula as reference
        if (d < d2) {
            if (d < d1) {
                d2 = d1; i2 = i1;
                if (d < d0) { d1 = d0; i1 = i0; d0 = d; i0 = s; }
                else        { d1 = d;  i1 = s; }
            } else          { d2 = d;  i2 = s; }
        }
    }

    const float r0 = 1.0f / fmaxf(d0, 1e-8f);
    const float r1 = 1.0f / fmaxf(d1, 1e-8f);
    const float r2 = 1.0f / fmaxf(d2, 1e-8f);
    const float inv = 1.0f / (r0 + r1 + r2);

    const size_t o = ((size_t)b * NN + n) * 3;
    idx_out[o] = i0; idx_out[o + 1] = i1; idx_out[o + 2] = i2;
    w_out[o] = r0 * inv; w_out[o + 1] = r1 * inv; w_out[o + 2] = r2 * inv;
}

// ---------------------------------------------------------------------------
// Kernel 2: 3-point inverse-distance interpolation -> (B, 256, N) channel-major
// ---------------------------------------------------------------------------
__global__ __launch_bounds__(256)
void interp_kernel(const float* __restrict__ fea2, const int* __restrict__ idx,
                   const float* __restrict__ wts, float* __restrict__ out)
{
    const int b = blockIdx.y;
    const int n = blockIdx.x * 256 + threadIdx.x;
    const size_t o = ((size_t)b * NN + n) * 3;
    const int i0 = idx[o], i1 = idx[o + 1], i2 = idx[o + 2];
    const float w0 = wts[o], w1 = wts[o + 1], w2 = wts[o + 2];

    const float* fb = fea2 + (size_t)b * D2C * SS;
    float* ob = out + (size_t)b * D2C * NN + n;
    #pragma unroll 4
    for (int c = 0; c < D2C; ++c) {
        const float* row = fb + (size_t)c * SS;
        ob[(size_t)c * NN] = w0 * row[i0] + w1 * row[i1] + w2 * row[i2];
    }
}

// ---------------------------------------------------------------------------
// Kernel 3/5: fused GEMM (Y = W @ X + bias), optional input act (BN+ReLU of the
// previous layer applied in the LDS loader).  Block = 256 threads = 8 waves,
// computes a 256(M) x 64(N) output tile.  Each wave: 2 M-tiles x 4 N-tiles of
// v_wmma_f32_16x16x4_f32 accumulators.
//   X[c][n] = Xa[b][c][n]       for c <  split  (layout (B, Ca, Npts))
//           = Xb[b][c-split][n] for c >= split  (layout (B, Cb, Npts))
// ---------------------------------------------------------------------------
__global__ __launch_bounds__(256)
void gemm_bn_kernel(const float* __restrict__ Xa, const float* __restrict__ Xb,
                    const float* __restrict__ W,  const float* __restrict__ bias,
                    const float* __restrict__ act_scale, const float* __restrict__ act_shift,
                    float* __restrict__ Y,
                    int K, int split, int Ca, int Cb, int Npts, int nblk_per_b)
{
    __shared__ float lds[16 * 72];                 // 16 K-rows x 64 cols, stride 72

    const int tid  = threadIdx.x;
    const int wave = tid >> 5;
    const int lane = tid & 31;
    const int ln   = lane & 15;
    const int hi   = lane >> 4;
    const int b    = blockIdx.x / nblk_per_b;
    const int n0   = (blockIdx.x % nblk_per_b) * 64;

    const int kl = tid >> 4;          // 0..15 : K-row this thread stages
    const int nl = (tid & 15) << 2;   // 0..60 : column (float4)

    v8f acc[2][4];
    #pragma unroll
    for (int i = 0; i < 2; ++i)
        #pragma unroll
        for (int t = 0; t < 4; ++t)
            #pragma unroll
            for (int j = 0; j < 8; ++j) acc[i][t][j] = 0.0f;

    const int r0 = wave << 5;         // first M row of this wave

    for (int k0 = 0; k0 < K; k0 += 16) {
        __syncthreads();
        // ---- stage 16 x 64 X-chunk into LDS (with optional BN+ReLU) ----
        const int c = k0 + kl;
        const float* src = (c < split)
            ? (Xa + ((size_t)b * Ca + c) * Npts)
            : (Xb + ((size_t)b * Cb + (c - split)) * Npts);
        float4 v = *(const float4*)(src + n0 + nl);
        if (act_scale) {
            const float s = act_scale[c], sh = act_shift[c];
            v.x = fmaxf(0.0f, fmaf(v.x, s, sh));
            v.y = fmaxf(0.0f, fmaf(v.y, s, sh));
            v.z = fmaxf(0.0f, fmaf(v.z, s, sh));
            v.w = fmaxf(0.0f, fmaf(v.w, s, sh));
        }
        *(float4*)(&lds[kl * 72 + nl]) = v;
        __syncthreads();

        // ---- 4 K-substeps of 4 ----
        #pragma unroll
        for (int s4 = 0; s4 < 4; ++s4) {
            const int k = k0 + (s4 << 2);
            // A fragments: v0 <-> K = k+2*hi, v1 <-> K = k+1+2*hi  (ISA 16x4 layout)
            const v2f a0 = *(const v2f*)(W + (size_t)(r0 + ln)      * K + k + 2 * hi);
            const v2f a1 = *(const v2f*)(W + (size_t)(r0 + 16 + ln) * K + k + 2 * hi);
            const int kb = (s4 << 2) + 2 * hi;
            #pragma unroll
            for (int t = 0; t < 4; ++t) {
                v2f bf;
                bf.x = lds[kb * 72       + t * 16 + ln];
                bf.y = lds[(kb + 1) * 72 + t * 16 + ln];
                acc[0][t] = WMMA_F32(a0, bf, acc[0][t]);
                acc[1][t] = WMMA_F32(a1, bf, acc[1][t]);
            }
        }
    }

    // ---- epilogue: D vgpr r, lanes0-15 -> M=r, lanes16-31 -> M=r+8 ----
    #pragma unroll
    for (int i = 0; i < 2; ++i) {
        const int cb = (wave << 5) + (i << 4) + (hi << 3);
        #pragma unroll
        for (int r = 0; r < 8; ++r) {
            const int ch = cb + r;
            const float bv = bias[ch];
            float* dst = Y + ((size_t)b * COUT + ch) * Npts + n0 + ln;
            #pragma unroll
            for (int t = 0; t < 4; ++t)
                dst[t * 16] = acc[i][t][r] + bv;
        }
    }
}

// ---------------------------------------------------------------------------
// Kernel 4/6: per-channel sum / sum-of-squares (one block per (b, c) row)
// ---------------------------------------------------------------------------
__global__ __launch_bounds__(256)
void stats_kernel(const float* __restrict__ Y, float* __restrict__ sums, int Npts)
{
    __shared__ float s1[256], s2[256];
    const int row = blockIdx.x;          // b*256 + c
    const int c = row & 255;
    const float* p = Y + (size_t)row * Npts;
    float a = 0.0f, q = 0.0f;
    for (int i = threadIdx.x; i < Npts; i += 256) {
        const float v = p[i];
        a += v; q += v * v;
    }
    s1[threadIdx.x] = a; s2[threadIdx.x] = q;
    __syncthreads();
    for (int st = 128; st > 0; st >>= 1) {
        if (threadIdx.x < st) {
            s1[threadIdx.x] += s1[threadIdx.x + st];
            s2[threadIdx.x] += s2[threadIdx.x + st];
        }
        __syncthreads();
    }
    if (threadIdx.x == 0) {
        atomicAdd(&sums[c], s1[0]);
        atomicAdd(&sums[256 + c], s2[0]);
    }
}

__global__ void finalize_kernel(const float* __restrict__ sums,
                                const float* __restrict__ gamma,
                                const float* __restrict__ beta,
                                float* __restrict__ sc, float* __restrict__ sh,
                                float inv_cnt)
{
    const int c = threadIdx.x;
    const float mean = sums[c] * inv_cnt;
    const float var  = sums[256 + c] * inv_cnt - mean * mean;
    const float s    = gamma[c] * rsqrtf(var + EPSB);
    sc[c] = s;
    sh[c] = beta[c] - mean * s;
}

// ---------------------------------------------------------------------------
// Kernel 7: in-place BN + ReLU on the final output
// ---------------------------------------------------------------------------
__global__ __launch_bounds__(256)
void apply_kernel(float* __restrict__ Y, const float* __restrict__ sc,
                  const float* __restrict__ sh, int Npts)
{
    const int i = blockIdx.x * 256 + threadIdx.x;
    const int c = (i / Npts) & 255;
    Y[i] = fmaxf(0.0f, fmaf(Y[i], sc[c], sh[c]));
}

// ---------------------------------------------------------------------------
extern "C" void kernel_launch(void* const* d_in, const int* in_sizes, int n_in,
                              void* d_out, int out_size, void* d_ws, size_t ws_size,
                              hipStream_t stream)
{
    const float* xyz1 = (const float*)d_in[0];
    const float* xyz2 = (const float*)d_in[1];
    const float* fea1 = (const float*)d_in[2];
    const float* fea2 = (const float*)d_in[3];
    const float* W0   = (const float*)d_in[4];
    const float* b0   = (const float*)d_in[5];
    const float* g0   = (const float*)d_in[6];
    const float* be0  = (const float*)d_in[7];
    const float* W1   = (const float*)d_in[8];
    const float* b1   = (const float*)d_in[9];
    const float* g1   = (const float*)d_in[10];
    const float* be1  = (const float*)d_in[11];
    float* out = (float*)d_out;

    // workspace layout (floats)
    float* ws = (float*)d_ws;
    const size_t NP = (size_t)BB * COUT * NN;         // 16.7M elems
    float* interp = ws;                                // (B,256,N)
    float* y0     = ws + NP;                           // (B,256,N) raw layer-0 output
    int*   idx    = (int*)(ws + 2 * NP);               // (B,N,3)
    float* wts    = ws + 2 * NP + (size_t)BB * NN * 3; // (B,N,3)
    float* stats0 = wts + (size_t)BB * NN * 3;         // 512
    float* sc0    = stats0 + 512;
    float* sh0    = sc0 + 256;
    float* stats1 = sh0 + 256;                         // 512
    float* sc1    = stats1 + 512;
    float* sh1    = sc1 + 256;

    hipMemsetAsync(stats0, 0, 512 * sizeof(float), stream);
    hipMemsetAsync(stats1, 0, 512 * sizeof(float), stream);

    const dim3 gPts(NN / 256, BB);
    knn3_kernel<<<gPts, 256, 0, stream>>>(xyz1, xyz2, idx, wts);
    interp_kernel<<<gPts, 256, 0, stream>>>(fea2, idx, wts, interp);

    const int nblk = NN / 64;
    const dim3 gG(BB * nblk);
    const float inv_cnt = 1.0f / (float)(BB * NN);

    // layer 0: X = concat(fea1, interp) along channels; no input activation
    gemm_bn_kernel<<<gG, 256, 0, stream>>>(fea1, interp, W0, b0, nullptr, nullptr,
                                           y0, C0K, D1C, D1C, D2C, NN, nblk);
    stats_kernel<<<BB * COUT, 256, 0, stream>>>(y0, stats0, NN);
    finalize_kernel<<<1, 256, 0, stream>>>(stats0, g0, be0, sc0, sh0, inv_cnt);

    // layer 1: BN0 + ReLU fused into the loader; raw y1 -> d_out
    gemm_bn_kernel<<<gG, 256, 0, stream>>>(y0, y0, W1, b1, sc0, sh0,
                                           out, COUT, COUT, COUT, COUT, NN, nblk);
    stats_kernel<<<BB * COUT, 256, 0, stream>>>(out, stats1, NN);
    finalize_kernel<<<1, 256, 0, stream>>>(stats1, g1, be1, sc1, sh1, inv_cnt);

    apply_kernel<<<(unsigned)(NP / 256), 256, 0, stream>>>(out, sc1, sh1, NN);
}